// PauliEnergy_27238682591801
// MI455X (gfx1250) — compile-verified
//
#include <hip/hip_runtime.h>
#include <math.h>

#define B_  128
#define S_  8000
#define HS  9
#define HHS 8
#define NH  4

typedef float v2f __attribute__((ext_vector_type(2)));
typedef float v8f __attribute__((ext_vector_type(8)));

// ---------------------------------------------------------------------------
// Kernel 1: x+pos_bias -> layernorm -> QKV projection (WMMA f32 16x16x4)
//           -> store q,v (token-major), per-block bind partial sums.
// One block = 128 threads (4 waves), 64 tokens of one batch.
// ---------------------------------------------------------------------------
__global__ void __launch_bounds__(128)
k1_proj(const float* __restrict__ x,  const float* __restrict__ pb,
        const float* __restrict__ lg, const float* __restrict__ lb,
        const float* __restrict__ Wq, const float* __restrict__ bq,
        const float* __restrict__ Wk, const float* __restrict__ bk,
        const float* __restrict__ Wv, const float* __restrict__ bv,
        float* __restrict__ q_ws, float* __restrict__ v_ws,
        float* __restrict__ bind_part)
{
    __shared__ float xln[64][13];    // K padded 9 -> 12 (cols 9..11 zero)
    __shared__ float Bls[12][32];    // B[k][n] = Wcat[n][k]; n: 0-7 Wq, 8-15 Wk, 16-23 Wv, 24-31 zero
    __shared__ float qkv[64][25];    // feat: 0-7 q, 8-15 k, 16-23 v
    __shared__ float bindv[64][8];

    const int tid   = threadIdx.x;
    const int b     = blockIdx.x / (S_ / 64);
    const int chunk = blockIdx.x % (S_ / 64);
    const int s0    = chunk * 64;

    // stage B matrix (tiny, L2-resident weights)
    for (int idx = tid; idx < 12 * 32; idx += 128) {
        int k = idx >> 5, n = idx & 31;
        float val = 0.f;
        if (k < HS) {
            if      (n < 8)  val = Wq[n * HS + k];
            else if (n < 16) val = Wk[(n - 8) * HS + k];
            else if (n < 24) val = Wv[(n - 16) * HS + k];
        }
        Bls[k][n] = val;
    }

    // layernorm: one thread per token
    if (tid < 64) {
        const int s = s0 + tid;
        const float* xr = x  + ((size_t)b * S_ + s) * HS;
        const float* pr = pb + (size_t)s * HS;
        float xv[HS]; float m = 0.f;
        #pragma unroll
        for (int j = 0; j < HS; j++) { xv[j] = xr[j] + pr[j]; m += xv[j]; }
        m *= (1.f / HS);
        float var = 0.f;
        #pragma unroll
        for (int j = 0; j < HS; j++) { float d = xv[j] - m; var += d * d; }
        var *= (1.f / HS);
        const float r = rsqrtf(var + 1e-6f);
        #pragma unroll
        for (int j = 0; j < HS; j++) xln[tid][j] = (xv[j] - m) * r * lg[j] + lb[j];
        xln[tid][9] = 0.f; xln[tid][10] = 0.f; xln[tid][11] = 0.f;
    }
    __syncthreads();

    // WMMA: wave w -> tokens [w*16, w*16+16).  D0: N=0..15 (q,k), D1: N=0..7 (v)
    const int lane  = tid & 31;
    const int w     = tid >> 5;
    const int nrow  = lane & 15;          // M for A, N for B
    const int khalf = (lane >> 4) * 2;    // 0 or 2
    v8f d0 = {0,0,0,0,0,0,0,0};
    v8f d1 = {0,0,0,0,0,0,0,0};
    #pragma unroll
    for (int kc = 0; kc < 12; kc += 4) {
        v2f a, bt0, bt1;
        a.x   = xln[w * 16 + nrow][kc + khalf];
        a.y   = xln[w * 16 + nrow][kc + khalf + 1];
        bt0.x = Bls[kc + khalf][nrow];
        bt0.y = Bls[kc + khalf + 1][nrow];
        bt1.x = Bls[kc + khalf][16 + nrow];
        bt1.y = Bls[kc + khalf + 1][16 + nrow];
        d0 = __builtin_amdgcn_wmma_f32_16x16x4_f32(false, a, false, bt0, (short)0, d0, false, false);
        d1 = __builtin_amdgcn_wmma_f32_16x16x4_f32(false, a, false, bt1, (short)0, d1, false, false);
    }
    // scatter D: VGPR j holds rows M=j (lanes 0-15) / M=j+8 (lanes 16-31), col N=lane&15
    const int trow = w * 16 + ((lane >> 4) << 3);
    #pragma unroll
    for (int j = 0; j < 8; j++) {
        qkv[trow + j][nrow] = d0[j];
        if (nrow < 8) qkv[trow + j][16 + nrow] = d1[j];
    }
    __syncthreads();

    // per-token: add biases, emit q/v, bind contributions
    if (tid < 64) {
        const int s = s0 + tid;
        float qv[8], kv[8], vv[8];
        #pragma unroll
        for (int j = 0; j < 8; j++) {
            qv[j] = qkv[tid][j]      + bq[j];
            kv[j] = qkv[tid][8 + j]  + bk[j];
            vv[j] = qkv[tid][16 + j] + bv[j];
        }
        float* qo = q_ws + ((size_t)b * S_ + s) * 8;
        float* vo = v_ws + ((size_t)b * S_ + s) * 8;
        #pragma unroll
        for (int j = 0; j < 8; j++) { qo[j] = qv[j]; vo[j] = vv[j]; }
        #pragma unroll
        for (int h = 0; h < NH; h++) {
            float k0 = kv[2*h], k1 = kv[2*h+1], v0 = vv[2*h], v1 = vv[2*h+1];
            bindv[tid][2*h]     = k0 * v0 + k1 * v1;   // circular conv, HD=2
            bindv[tid][2*h + 1] = k0 * v1 + k1 * v0;
        }
    }
    __syncthreads();

    // deterministic in-block reduction (fixed order), one partial per block
    if (tid < 8) {
        float s = 0.f;
        for (int t = 0; t < 64; t++) s += bindv[t][tid];
        bind_part[((size_t)b * (S_ / 64) + chunk) * 8 + tid] = s;
    }
}

// Deterministic reduction of bind partials over 125 chunks per batch
__global__ void k1b_reduce(const float* __restrict__ part, float* __restrict__ bind)
{
    int t = blockIdx.x * blockDim.x + threadIdx.x;   // 0..B*NH*2-1
    if (t >= B_ * NH * 2) return;
    int b = t >> 3, i = t & 7;
    float s = 0.f;
    for (int c = 0; c < S_ / 64; c++) s += part[((size_t)b * (S_ / 64) + c) * 8 + i];
    bind[t] = s;
}

// ---------------------------------------------------------------------------
// Kernel 2: vp = unbind(bind,q); scale = cos(v,vp) + mask; e = exp(scale);
// per-block deterministic tree sum -> se_part.
// ---------------------------------------------------------------------------
__global__ void __launch_bounds__(256)
k2_scale(const float* __restrict__ q_ws, const float* __restrict__ v_ws,
         const float* __restrict__ bind, const float* __restrict__ mask,
         float* __restrict__ e_ws, float* __restrict__ se_part)
{
    __shared__ float red[256];
    const int tid = threadIdx.x;
    const int bh  = blockIdx.x >> 5;               // 32 blocks per (b,h)
    const int blk = blockIdx.x & 31;
    const int s   = blk * 256 + tid;
    const int b   = bh >> 2, h = bh & 3;

    float e = 0.f;
    if (s < S_) {
        const float c0 = bind[bh * 2], c1 = bind[bh * 2 + 1];
        const size_t tb = ((size_t)b * S_ + s) * 8 + 2 * h;
        const float q0 = q_ws[tb], q1 = q_ws[tb + 1];
        const float v0 = v_ws[tb], v1 = v_ws[tb + 1];
        const float vp0 = c0 * q0 + c1 * q1;       // circular correlation, HD=2
        const float vp1 = c0 * q1 + c1 * q0;
        const float num = v0 * vp0 + v1 * vp1;
        const float na  = fmaxf(sqrtf(v0 * v0 + v1 * v1), 1e-8f);
        const float nb  = fmaxf(sqrtf(vp0 * vp0 + vp1 * vp1), 1e-8f);
        float sc = num / (na * nb);
        sc += (1.f - mask[(size_t)bh * S_ + s]) * (-1e9f);
        e = expf(sc);                               // sc <= 1: no max-shift needed
        e_ws[(size_t)bh * S_ + s] = e;
    }
    red[tid] = e;
    __syncthreads();
    for (int off = 128; off > 0; off >>= 1) {
        if (tid < off) red[tid] += red[tid + off];
        __syncthreads();
    }
    if (tid == 0) se_part[(size_t)bh * 32 + blk] = red[0];
}

__global__ void k2b_reduce(const float* __restrict__ part, float* __restrict__ sumexp)
{
    int t = blockIdx.x * blockDim.x + threadIdx.x;  // 0..B*NH-1
    if (t >= B_ * NH) return;
    float s = 0.f;
    for (int c = 0; c < 32; c++) s += part[(size_t)t * 32 + c];
    sumexp[t] = s;
}

// ---------------------------------------------------------------------------
// Kernel 3: weight = e/sum; attn = weight*v; gelu(attn@W2^T+b2)+x_ln; @Wo+bo.
// LayerNorm recomputed (cheaper than storing 37MB of x_ln).
// ---------------------------------------------------------------------------
__global__ void __launch_bounds__(256)
k3_out(const float* __restrict__ x,  const float* __restrict__ pb,
       const float* __restrict__ lg, const float* __restrict__ lb,
       const float* __restrict__ v_ws, const float* __restrict__ e_ws,
       const float* __restrict__ sumexp,
       const float* __restrict__ W2, const float* __restrict__ b2,
       const float* __restrict__ Wo, const float* __restrict__ bo,
       float* __restrict__ out)
{
    const int tid = threadIdx.x;
    const int b   = blockIdx.x >> 5;
    const int s   = (blockIdx.x & 31) * 256 + tid;
    if (s >= S_) return;

    const float* xr = x  + ((size_t)b * S_ + s) * HS;
    const float* pr = pb + (size_t)s * HS;
    float xv[HS]; float m = 0.f;
    #pragma unroll
    for (int j = 0; j < HS; j++) { xv[j] = xr[j] + pr[j]; m += xv[j]; }
    m *= (1.f / HS);
    float var = 0.f;
    #pragma unroll
    for (int j = 0; j < HS; j++) { float d = xv[j] - m; var += d * d; }
    var *= (1.f / HS);
    const float r = rsqrtf(var + 1e-6f);
    float xln[HS];
    #pragma unroll
    for (int j = 0; j < HS; j++) xln[j] = (xv[j] - m) * r * lg[j] + lb[j];

    float attn[8];
    const float* vr = v_ws + ((size_t)b * S_ + s) * 8;
    #pragma unroll
    for (int h = 0; h < NH; h++) {
        const float wgt = e_ws[((size_t)(b * NH + h)) * S_ + s] / sumexp[b * NH + h];
        attn[2*h]     = wgt * vr[2*h];
        attn[2*h + 1] = wgt * vr[2*h + 1];
    }

    float o = bo[0];
    #pragma unroll
    for (int i = 0; i < HS; i++) {
        float z = b2[i];
        #pragma unroll
        for (int j = 0; j < 8; j++) z += attn[j] * W2[i * 8 + j];
        const float g = 0.5f * z * (1.f + erff(z * 0.70710678118654752f)); // exact gelu
        o += (g + xln[i]) * Wo[i];
    }
    out[(size_t)b * S_ + s] = o;
}

// ---------------------------------------------------------------------------
extern "C" void kernel_launch(void* const* d_in, const int* in_sizes, int n_in,
                              void* d_out, int out_size, void* d_ws, size_t ws_size,
                              hipStream_t stream)
{
    const float* x    = (const float*)d_in[0];
    const float* mask = (const float*)d_in[1];
    const float* pb   = (const float*)d_in[2];
    const float* lg   = (const float*)d_in[3];
    const float* lb   = (const float*)d_in[4];
    const float* Wq   = (const float*)d_in[5];
    const float* bq   = (const float*)d_in[6];
    const float* Wk   = (const float*)d_in[7];
    const float* bk   = (const float*)d_in[8];
    const float* Wv   = (const float*)d_in[9];
    const float* bv   = (const float*)d_in[10];
    // d_in[11]=W1, d_in[12]=b1 : dead code in reference
    const float* W2   = (const float*)d_in[13];
    const float* b2   = (const float*)d_in[14];
    const float* Wo   = (const float*)d_in[15];
    const float* bo   = (const float*)d_in[16];
    float* out = (float*)d_out;

    float* q_ws      = (float*)d_ws;                       // B*S*8
    float* v_ws      = q_ws + (size_t)B_ * S_ * 8;         // B*S*8
    float* e_ws      = v_ws + (size_t)B_ * S_ * 8;         // B*NH*S
    float* bind_part = e_ws + (size_t)B_ * NH * S_;        // B*125*8
    float* se_part   = bind_part + (size_t)B_ * (S_/64) * 8; // B*NH*32
    float* bind      = se_part + (size_t)B_ * NH * 32;     // B*NH*2
    float* sumexp    = bind + B_ * NH * 2;                 // B*NH

    k1_proj <<<B_ * (S_ / 64), 128, 0, stream>>>(x, pb, lg, lb, Wq, bq, Wk, bk, Wv, bv,
                                                 q_ws, v_ws, bind_part);
    k1b_reduce<<<(B_ * NH * 2 + 255) / 256, 256, 0, stream>>>(bind_part, bind);
    k2_scale<<<B_ * NH * 32, 256, 0, stream>>>(q_ws, v_ws, bind, mask, e_ws, se_part);
    k2b_reduce<<<(B_ * NH + 255) / 256, 256, 0, stream>>>(se_part, sumexp);
    k3_out  <<<B_ * 32, 256, 0, stream>>>(x, pb, lg, lb, v_ws, e_ws, sumexp,
                                          W2, b2, Wo, bo, out);
}